// EntropyDown_38285338476634
// MI455X (gfx1250) — compile-verified
//
#include <hip/hip_runtime.h>
#include <hip/hip_bf16.h>

typedef __attribute__((ext_vector_type(2))) float v2f;
typedef __attribute__((ext_vector_type(8))) float v8f;

// Problem constants (match reference setup_inputs / RATIO=4)
#define B_  4
#define L_  4096
#define C_  512
#define NH_ 8
#define S_  1024
#define K_  (L_ / 4)        // 1024
#define CH_ (C_ / NH_)      // 64
#define NROWS_ (B_ * NH_ * L_)  // 131072 entropy rows

// ---------------------------------------------------------------------------
// Kernel 1: neg_entropy[row] = sum_s exp(attn[row,s]) * attn[row,s]
// One wave32 per row of S_=1024 floats; 8 waves (256 threads) per block.
// Cross-lane reduction via V_WMMA_F32_16X16X4_F32 (exact fp32 reduce).
// ---------------------------------------------------------------------------
__global__ __launch_bounds__(256) void entropy_kernel(
    const float* __restrict__ attn, float* __restrict__ ne) {
  const int wave = threadIdx.x >> 5;
  const int lane = threadIdx.x & 31;
  const int row  = blockIdx.x * 8 + wave;

  const float4* p = (const float4*)(attn + (size_t)row * S_) + lane;
  float acc = 0.f;
#pragma unroll
  for (int i = 0; i < S_ / (4 * 32); ++i) {   // 8 iterations of float4
    float4 v = p[i * 32];
    acc += __expf(v.x) * v.x;
    acc += __expf(v.y) * v.y;
    acc += __expf(v.z) * v.z;
    acc += __expf(v.w) * v.w;
  }

  // WMMA fp32 cross-lane reduction:
  // A (16x4 fp32, 2 VGPRs): lanes 0-15 -> A[m,0]=acc, lanes 16-31 -> A[m,2]=acc,
  // second VGPR = 0. B = all ones => D[m,n] = p_m + p_{m+16} (same for all n).
  v2f a; a.x = acc; a.y = 0.f;
  v2f b; b.x = 1.f; b.y = 1.f;
  v8f c = {};
  v8f d = __builtin_amdgcn_wmma_f32_16x16x4_f32(
      /*neg_a=*/false, a, /*neg_b=*/false, b,
      /*c_mod=*/(short)0, c, /*reuse_a=*/false, /*reuse_b=*/false);

  // lanes 0-15 hold rows m=0..7 in d[0..7]; lanes 16-31 hold rows m=8..15.
  float s = d[0] + d[1] + d[2] + d[3] + d[4] + d[5] + d[6] + d[7];
  float tot = s + __shfl_xor(s, 16, 32);  // all lanes end with the full sum
  if (lane == 0) ne[row] = tot;
}

// ---------------------------------------------------------------------------
// Kernel 2: per (b,h) row of L_=4096 values, stable descending bitonic sort of
// (value, index) pairs in LDS; emit indices of the top K_ entries in order.
// Comparator: value desc, index asc (matches JAX top_k ordering).
// ---------------------------------------------------------------------------
__global__ __launch_bounds__(1024) void topk_kernel(
    const float* __restrict__ ne, int* __restrict__ idx) {
  __shared__ float sval[L_];
  __shared__ int   sidx[L_];
  const int row = blockIdx.x;  // b*NH_ + h
  const float* nrow = ne + (size_t)row * L_;

  for (int i = threadIdx.x; i < L_; i += 1024) {
    sval[i] = nrow[i];
    sidx[i] = i;
  }
  __syncthreads();

  for (int ksz = 2; ksz <= L_; ksz <<= 1) {
    for (int j = ksz >> 1; j > 0; j >>= 1) {
      for (int i = threadIdx.x; i < L_; i += 1024) {
        int ixj = i ^ j;
        if (ixj > i) {
          float v0 = sval[i], v1 = sval[ixj];
          int   i0 = sidx[i], i1 = sidx[ixj];
          // "elem[ixj] belongs before elem[i]" in descending order:
          bool before = (v1 > v0) || (v1 == v0 && i1 < i0);
          bool desc = ((i & ksz) == 0);
          if (before == desc) {
            sval[i] = v1; sval[ixj] = v0;
            sidx[i] = i1; sidx[ixj] = i0;
          }
        }
      }
      __syncthreads();
    }
  }

  for (int i = threadIdx.x; i < K_; i += 1024)
    idx[(size_t)row * K_ + i] = sidx[i];
}

// ---------------------------------------------------------------------------
// Kernel 3: gather. out layout = [x_out (B,K,C) ; coord_out (B,K,C)] flat.
// out[b,j, h*CH + c] = src[b, idx[b,h,j], h*CH + c]. float4 per thread.
// ---------------------------------------------------------------------------
__global__ __launch_bounds__(256) void gather_kernel(
    const float4* __restrict__ x, const float4* __restrict__ coord,
    const int* __restrict__ idx, float4* __restrict__ out) {
  const int C4 = C_ / 4;                 // 128 float4 per token row
  const int half4 = B_ * K_ * C4;       // float4 count of one output tensor
  int t = blockIdx.x * 256 + threadIdx.x;

  const float4* src = x;
  int r = t;
  if (t >= half4) { src = coord; r = t - half4; }

  int b  = r / (K_ * C4);
  int r2 = r - b * (K_ * C4);
  int j  = r2 >> 7;            // / C4
  int c4 = r2 & (C4 - 1);
  int h  = c4 >> 4;            // / (CH_/4)
  int token = idx[(b * NH_ + h) * K_ + j];
  out[t] = src[((size_t)b * L_ + token) * C4 + c4];
}

// ---------------------------------------------------------------------------
extern "C" void kernel_launch(void* const* d_in, const int* in_sizes, int n_in,
                              void* d_out, int out_size, void* d_ws, size_t ws_size,
                              hipStream_t stream) {
  const float* x     = (const float*)d_in[0];
  const float* coord = (const float*)d_in[1];
  const float* attn  = (const float*)d_in[2];
  float* out = (float*)d_out;

  // workspace: neg_entropy [NROWS_] floats, then idx [B_*NH_*K_] ints
  float* ne  = (float*)d_ws;
  int*   idx = (int*)((char*)d_ws + (size_t)NROWS_ * sizeof(float));

  // 1) entropy reduction: 131072 rows, 8 rows per 256-thread block
  entropy_kernel<<<NROWS_ / 8, 256, 0, stream>>>(attn, ne);

  // 2) top-k per (b,h): 32 workgroups of 1024 threads
  topk_kernel<<<B_ * NH_, 1024, 0, stream>>>(ne, idx);

  // 3) gather: 2 * B_*K_*(C_/4) float4 elements
  const int total4 = 2 * B_ * K_ * (C_ / 4);
  gather_kernel<<<total4 / 256, 256, 0, stream>>>(
      (const float4*)x, (const float4*)coord, idx, (float4*)out);
}